// IoUAssigner_72353019068752
// MI455X (gfx1250) — compile-verified
//
#include <hip/hip_runtime.h>
#include <hip/hip_bf16.h>
#include <cstdint>

#define MAX_GTS   128
#define BLOCK     256
#define POS_THR   0.7f
#define NEG_THR   0.3f
#define MINPOS    0.3f

typedef unsigned int u32x4 __attribute__((ext_vector_type(4)));
typedef int          i32x4 __attribute__((ext_vector_type(4)));
typedef int          i32x8 __attribute__((ext_vector_type(8)));

__device__ __forceinline__ float fast_rcp(float x) {
#if __has_builtin(__builtin_amdgcn_rcpf)
  return __builtin_amdgcn_rcpf(x);
#else
  return 1.0f / x;
#endif
}

// ---------------------------------------------------------------------------
// Kernel 0: zero the per-GT packed (iou, anchor) reduction slots in d_ws.
// Runs every launch so graph replays are deterministic.
// ---------------------------------------------------------------------------
__global__ void iou_init_kernel(unsigned long long* __restrict__ gkeys, int M) {
  int t = blockIdx.x * blockDim.x + threadIdx.x;
  if (t < M) gkeys[t] = 0ull;
}

// ---------------------------------------------------------------------------
// Kernel 1: per-anchor pass.
//   - TDM (tensor_load_to_lds) stages the raw GT tile (M*6 f32) into LDS.
//   - Threads 0..M-1 repack to 32B records [l0,l1,l2,r0 | r1,r2,area,0] so the
//     hot loop needs only two ds_load_b128 broadcasts per GT.
//   - Each thread computes IoU vs all GTs, tracks (max, first-argmax), writes
//     labels/boxes assuming no override, and feeds the per-GT column max via
//     packed-u64 ds_max atomics (only when iou >= MINPOS), flushed once per
//     block with a global atomicMax.
// ---------------------------------------------------------------------------
__global__ __launch_bounds__(BLOCK) void iou_main_kernel(
    const float* __restrict__ boxes,
    const float* __restrict__ gts,
    const int* __restrict__ gt_labels,
    int M, int N,
    int* __restrict__ labels_out,
    float* __restrict__ boxes_out,
    unsigned long long* __restrict__ gkeys)
{
  __shared__ float  sRaw[MAX_GTS * 6];            // TDM destination (AoS x6)
  __shared__ float4 sGT[MAX_GTS][2];              // repacked 32B records
  __shared__ unsigned long long sKey[MAX_GTS];

  const int tid = threadIdx.x;
  const int K   = M * 6;                          // f32 elements in GT tile

  if (tid < MAX_GTS) sKey[tid] = 0ull;

  // ---- Stage GT boxes into LDS via the Tensor Data Mover (wave 0) ----
#if __has_builtin(__builtin_amdgcn_tensor_load_to_lds)
  if (tid < 32) {
    unsigned long long ga = (unsigned long long)(size_t)gts;
    unsigned lds_addr = (unsigned)(size_t)(&sRaw[0]);  // AS(3) offset bits

    u32x4 g0;
    g0.x = 1u;                                         // count = 1 valid D#
    g0.y = lds_addr;                                   // lds_addr
    g0.z = (unsigned)ga;                               // global_addr[31:0]
    g0.w = (unsigned)((ga >> 32) & 0x01FFFFFFull)      // global_addr[56:32]
         | 0x80000000u;                                // type = 2 ("image")

    i32x8 g1;
    g1[0] = (int)(2u << 16);                           // data_size = 4 bytes
    g1[1] = (int)(((unsigned)K & 0xFFFFu) << 16);      // tensor_dim0 lo16
    g1[2] = (int)((((unsigned)K >> 16) & 0xFFFFu)      // tensor_dim0 hi16
         | (1u << 16));                                // tensor_dim1 = 1
    g1[3] = (int)(((unsigned)K & 0xFFFFu) << 16);      // tile_dim0 = K
    g1[4] = 0;                                         // tile_dim1/2 unused
    g1[5] = K;                                         // tensor_dim0_stride
    g1[6] = 0;
    g1[7] = 0;

    i32x4 gz = {0, 0, 0, 0};
#if defined(__clang_major__) && (__clang_major__ >= 23)
#pragma message("CDNA5: using 6-arg __builtin_amdgcn_tensor_load_to_lds")
    i32x8 gz8 = {0, 0, 0, 0, 0, 0, 0, 0};
    __builtin_amdgcn_tensor_load_to_lds(g0, g1, gz, gz, gz8, 0);
#else
#pragma message("CDNA5: using 5-arg __builtin_amdgcn_tensor_load_to_lds")
    __builtin_amdgcn_tensor_load_to_lds(g0, g1, gz, gz, 0);
#endif
#if __has_builtin(__builtin_amdgcn_s_wait_tensorcnt)
    __builtin_amdgcn_s_wait_tensorcnt(0);
#else
    asm volatile("s_wait_tensorcnt 0" ::: "memory");
#endif
  }
#else
#pragma message("CDNA5: tensor_load_to_lds builtin NOT available - sync fallback")
  for (int t = tid; t < K; t += BLOCK) sRaw[t] = gts[t];
#endif
  __syncthreads();

  // Repack to 32-byte records: two b128 LDS broadcasts per GT in the hot loop.
  if (tid < M) {
    const float* g = &sRaw[tid * 6];
    const float a  = (g[3] - g[0]) * (g[4] - g[1]) * (g[5] - g[2]);
    sGT[tid][0] = make_float4(g[0], g[1], g[2], g[3]);
    sGT[tid][1] = make_float4(g[4], g[5], a, 0.0f);
  }
  __syncthreads();

  const int  i      = blockIdx.x * BLOCK + tid;
  const bool active = (i < N);
  const int  ii     = active ? i : (N - 1);

  // 6 f32 per box, 24B stride -> 8B aligned: three float2 loads.
  const float2* bp = (const float2*)(boxes + (size_t)ii * 6);
  const float2 p0 = bp[0], p1 = bp[1], p2 = bp[2];
  const float al0 = p0.x, al1 = p0.y, al2 = p1.x;
  const float ar0 = p1.y, ar1 = p2.x, ar2 = p2.y;
  const float areaA = (ar0 - al0) * (ar1 - al1) * (ar2 - al2);

  float best  = -1.0f;   // strict '>' keeps first index (jnp.argmax semantics)
  int   bestj = 0;
  const unsigned long long idxkey =
      0xFFFFFFFFull - (unsigned long long)(unsigned)i;  // ties -> lowest anchor

#pragma unroll 4
  for (int j = 0; j < M; ++j) {
    const float4 q0 = sGT[j][0];    // l0,l1,l2,r0
    const float4 q1 = sGT[j][1];    // r1,r2,area,0
    float ix = fminf(ar0, q0.w) - fmaxf(al0, q0.x);
    float iy = fminf(ar1, q1.x) - fmaxf(al1, q0.y);
    float iz = fminf(ar2, q1.y) - fmaxf(al2, q0.z);
    ix = fmaxf(ix, 0.0f);
    iy = fmaxf(iy, 0.0f);
    iz = fmaxf(iz, 0.0f);
    const float inter = ix * iy * iz;
    const float denom = areaA + q1.z - inter + 1e-7f;
    const float iou   = inter * fast_rcp(denom);

    if (iou > best) { best = iou; bestj = j; }

    // Only candidates with iou >= MINPOS can influence the low-quality
    // override; this keeps the per-GT column max exact whenever it matters
    // and makes the atomic path essentially free for random boxes.
    if (active && iou >= MINPOS) {
      unsigned long long key =
          ((unsigned long long)__float_as_uint(iou) << 32) | idxkey;
      atomicMax(&sKey[j], key);                       // ds_max_rtn_u64
    }
  }

  if (active) {
    int assigned;
    if (best >= POS_THR)      assigned = bestj + 1;
    else if (best < NEG_THR)  assigned = 0;
    else                      assigned = -1;

    float o0, o1, o2, o3, o4, o5;
    int lab;
    if (assigned > 0) {
      const float4 q0 = sGT[bestj][0];
      const float4 q1 = sGT[bestj][1];
      o0 = q0.x; o1 = q0.y; o2 = q0.z; o3 = q0.w; o4 = q1.x; o5 = q1.y;
      lab = gt_labels[bestj];
    } else {
      o0 = o1 = o2 = o3 = o4 = o5 = -1.0f;
      lab = (assigned == 0) ? 0 : -1;
    }
    labels_out[i] = lab;
    float2* op = (float2*)(boxes_out + (size_t)i * 6);
    op[0] = make_float2(o0, o1);
    op[1] = make_float2(o2, o3);
    op[2] = make_float2(o4, o5);
  }

  __syncthreads();
  // One global atomic per GT per block, only if this block saw a candidate.
  if (tid < M) {
    unsigned long long k = sKey[tid];
    if (k) atomicMax(&gkeys[tid], k);
  }
}

// ---------------------------------------------------------------------------
// Kernel 2: low-quality match override. For each GT with column max >= MINPOS,
// the largest GT index pointing at a given anchor wins (cand = i+1, .max()).
// ---------------------------------------------------------------------------
__global__ __launch_bounds__(MAX_GTS) void iou_patch_kernel(
    const float* __restrict__ gts,
    const int* __restrict__ gt_labels,
    int M,
    int* __restrict__ labels_out,
    float* __restrict__ boxes_out,
    const unsigned long long* __restrict__ gkeys)
{
  __shared__ unsigned long long sk[MAX_GTS];
  const int t = threadIdx.x;
  sk[t] = (t < M) ? gkeys[t] : 0ull;
  __syncthreads();

  if (t < M && sk[t] != 0ull) {   // nonzero key implies column max >= MINPOS
    const unsigned b = 0xFFFFFFFFu - (unsigned)(sk[t] & 0xFFFFFFFFull);
    bool win = true;
    for (int j = t + 1; j < M; ++j) {       // larger GT index overrides
      if (sk[j] != 0ull) {
        const unsigned b2 = 0xFFFFFFFFu - (unsigned)(sk[j] & 0xFFFFFFFFull);
        if (b2 == b) { win = false; break; }
      }
    }
    if (win) {
      labels_out[b] = gt_labels[t];
      const float* g  = &gts[t * 6];
      float*       ob = &boxes_out[(size_t)b * 6];
      #pragma unroll
      for (int c = 0; c < 6; ++c) ob[c] = g[c];
    }
  }
}

// ---------------------------------------------------------------------------
extern "C" void kernel_launch(void* const* d_in, const int* in_sizes, int n_in,
                              void* d_out, int out_size, void* d_ws, size_t ws_size,
                              hipStream_t stream) {
  const float* boxes     = (const float*)d_in[0];   // [N, 6] f32
  const float* gts       = (const float*)d_in[1];   // [M, 6] f32
  const int*   gt_labels = (const int*)d_in[2];     // [M]    i32

  const int N = in_sizes[0] / 6;
  int M = in_sizes[1] / 6;
  if (M > MAX_GTS) M = MAX_GTS;

  // d_out: [labels int32 x N][boxes f32 x 6N], concatenated in return order.
  int*   labels_out = (int*)d_out;
  float* boxes_out  = (float*)d_out + (size_t)N;

  unsigned long long* gkeys = (unsigned long long*)d_ws;  // 128 * u64

  iou_init_kernel<<<1, 256, 0, stream>>>(gkeys, M);

  const int nblocks = (N + BLOCK - 1) / BLOCK;
  iou_main_kernel<<<nblocks, BLOCK, 0, stream>>>(
      boxes, gts, gt_labels, M, N, labels_out, boxes_out, gkeys);

  iou_patch_kernel<<<1, MAX_GTS, 0, stream>>>(
      gts, gt_labels, M, labels_out, boxes_out, gkeys);
}